// SPDNormalization_20959440405164
// MI455X (gfx1250) — compile-verified
//
#include <hip/hip_runtime.h>

// ---------------------------------------------------------------------------
// SPD normalization on MI455X (gfx1250, wave32, WMMA f32 16x16x4).
// All 64x64 matrix functions are built from LDS-resident WMMA GEMMs:
//   sqrtm / inv-sqrtm : coupled Newton-Schulz
//   matlog            : inverse scaling-and-squaring + Gregory series
//   matexp            : scaling-and-squaring + Horner Taylor
// ---------------------------------------------------------------------------

#define LDA   66                 // padded LDS stride (bank-conflict free, 8B aligned)
#define BUF   (64 * LDA)         // one 64x64 fp32 matrix in LDS (padded)
#define NPART 64                 // partial accumulators for the tangent mean

#define WS_M    0                // mean matrix          (4096 f32)
#define WS_C    4096             // c   = sqrtm(mean)
#define WS_CI   8192             // c^-1
#define WS_C2I  12288            // final center_sqrt^-1
#define WS_PART 16384            // NPART x 4096 partial tangent sums
#define WS_TOTAL (WS_PART + NPART * 4096)

#define NS_PREP   14             // Newton-Schulz iters (mean sqrt)
#define NS_ROOT   7              // Newton-Schulz iters per matlog square root
#define K_ROOTS   4              // square roots before log series
#define K_EXP     6              // squarings in matexp
#define NS_FINAL  14             // Newton-Schulz iters (final sqrt)

typedef __attribute__((ext_vector_type(2))) float v2f;
typedef __attribute__((ext_vector_type(8))) float v8f;

// D = A*B over 64x64 fp32 LDS matrices; 128 threads = 4 waves, wave w owns
// rows [16w,16w+16), 4 column tiles, K swept in steps of 4 via WMMA f32.
// MODE 0: D = A*B      MODE 1: D = 1.5*I - 0.5*(A*B)     MODE 3: D = a*(A*B)+I
template <int MODE>
__device__ __forceinline__ void mm64(const float* A, const float* B, float* D,
                                     float alpha = 1.0f) {
  (void)alpha;
  __syncthreads();
  const int tid  = threadIdx.x;
  const int lane = tid & 31;
  const int wave = tid >> 5;
  const int m0   = wave << 4;
  const int lr   = lane & 15;
  const int hi   = lane >> 4;

  const v8f vz = {0.f, 0.f, 0.f, 0.f, 0.f, 0.f, 0.f, 0.f};
  v8f acc[4];
#pragma unroll
  for (int t = 0; t < 4; ++t) acc[t] = vz;

#pragma unroll 4
  for (int k0 = 0; k0 < 64; k0 += 4) {
    // A fragment (16x4): lanes 0-15 carry K={k0,k0+1}, lanes 16-31 K={k0+2,k0+3}
    const float* ap = A + (m0 + lr) * LDA + k0 + (hi << 1);
    v2f a;
    a.x = ap[0];
    a.y = ap[1];
    const float* b0 = B + (k0 + hi) * LDA;   // B fragment rows (4x16 tile)
    const float* b1 = b0 + 2 * LDA;
#pragma unroll
    for (int t = 0; t < 4; ++t) {
      const int n = (t << 4) + lr;
      v2f b;
      b.x = b0[n];
      b.y = b1[n];
      acc[t] = __builtin_amdgcn_wmma_f32_16x16x4_f32(
          false, a, false, b, (short)0, acc[t], false, false);
    }
  }

  // C/D layout: VGPR i -> row m0+i (lanes 0-15) / m0+8+i (lanes 16-31)
  const int mr = m0 + (hi << 3);
#pragma unroll
  for (int t = 0; t < 4; ++t) {
    const int col = (t << 4) + lr;
#pragma unroll
    for (int i = 0; i < 8; ++i) {
      float v = acc[t][i];
      const int row = mr + i;
      if (MODE == 1) v = ((row == col) ? 1.5f : 0.0f) - 0.5f * v;
      if (MODE == 3) v = alpha * v + ((row == col) ? 1.0f : 0.0f);
      D[row * LDA + col] = v;
    }
  }
  __syncthreads();
}

// D = a*S + b*I (elementwise; thread-owned element mapping, no sync needed
// between back-to-back ew_* calls)
__device__ __forceinline__ void ew_affine(float* D, const float* S, float a,
                                          float b) {
  for (int e = threadIdx.x; e < 4096; e += 128) {
    const int r = e >> 6, c = e & 63;
    D[r * LDA + c] = a * S[r * LDA + c] + ((r == c) ? b : 0.0f);
  }
}

__device__ __forceinline__ void set_identity(float* D) {
  for (int e = threadIdx.x; e < 4096; e += 128) {
    const int r = e >> 6, c = e & 63;
    D[r * LDA + c] = (r == c) ? 1.0f : 0.0f;
  }
}

__device__ __forceinline__ void g2l(float* dst, const float* __restrict__ src) {
  for (int i = threadIdx.x; i < 2048; i += 128) {
    const int e = i << 1;
    const int r = e >> 6, c = e & 63;
    const float2 v = reinterpret_cast<const float2*>(src)[i];
    *reinterpret_cast<float2*>(dst + r * LDA + c) = v;
  }
}

__device__ __forceinline__ void l2g(float* __restrict__ dst, const float* src) {
  for (int i = threadIdx.x; i < 2048; i += 128) {
    const int e = i << 1;
    const int r = e >> 6, c = e & 63;
    reinterpret_cast<float2*>(dst)[i] =
        *reinterpret_cast<const float2*>(src + r * LDA + c);
  }
}

__device__ __forceinline__ float trace64(const float* M, float* red) {
  __syncthreads();
  if (threadIdx.x < 64) red[threadIdx.x] = M[threadIdx.x * LDA + threadIdx.x];
  __syncthreads();
  if (threadIdx.x == 0) {
    float s = 0.f;
    for (int i = 0; i < 64; ++i) s += red[i];
    red[64] = s;
  }
  __syncthreads();
  const float t = red[64];
  __syncthreads();
  return t;
}

// Coupled Newton-Schulz: on exit pY = sqrt(Y0), pZ = inv(sqrt(Y0)).
// 4-buffer scheme: newZ is written into the retired Y buffer each iteration.
__device__ __forceinline__ void ns_sqrt(float*& pY, float*& pZ, float*& pT,
                                        float*& pU, int iters) {
  set_identity(pZ);
  for (int it = 0; it < iters; ++it) {
    mm64<1>(pZ, pY, pT);   // T = 1.5I - 0.5*Z*Y
    mm64<0>(pY, pT, pU);   // newY
    mm64<0>(pT, pZ, pY);   // newZ (old Y retired)
    float* t0 = pY;
    pY = pU;
    pU = pZ;
    pZ = t0;
  }
}

// ------------------------------- kernels -----------------------------------

__global__ void kzero(float* ws, int n) {
  const int i = blockIdx.x * blockDim.x + threadIdx.x;
  if (i < n) ws[i] = 0.0f;
}

// mean over batch: grid (32 element-chunks, 64 batch-chunks), fp32 atomics
__global__ void __launch_bounds__(128) kmean(const float* __restrict__ X,
                                             float* ws) {
  const int e  = blockIdx.x * 128 + threadIdx.x;   // element in [0,4096)
  const int b0 = blockIdx.y * 128;                 // batch chunk
  float s = 0.0f;
#pragma unroll 4
  for (int b = 0; b < 128; ++b) s += X[(size_t)(b0 + b) * 4096 + e];
  atomicAdd(&ws[WS_M + e], s * (1.0f / 8192.0f));
}

// single block: c = sqrtm(mean), c_inv = inv(c) via Newton-Schulz
__global__ void __launch_bounds__(128) kprep(float* ws) {
  __shared__ float sm[4 * BUF + 80];
  float* B0 = sm;
  float* B1 = sm + BUF;
  float* B2 = sm + 2 * BUF;
  float* B3 = sm + 3 * BUF;
  float* red = sm + 4 * BUF;

  g2l(B3, ws + WS_M);
  __syncthreads();
  const float s = trace64(B3, red) * (1.0f / 64.0f);   // spectral-scale proxy
  ew_affine(B3, B3, 1.0f / s, 0.0f);
  float *pY = B3, *pZ = B2, *pT = B1, *pU = B0;
  ns_sqrt(pY, pZ, pT, pU, NS_PREP);
  const float sq = sqrtf(s);
  ew_affine(pT, pY, sq, 0.0f);        // c
  ew_affine(pU, pZ, 1.0f / sq, 0.0f); // c_inv
  __syncthreads();
  l2g(ws + WS_C, pT);
  l2g(ws + WS_CI, pU);
}

// one block per batch element: P_b = c @ log(c^-1 X_b c^-1) @ c, accumulated
__global__ void __launch_bounds__(128) klog(const float* __restrict__ X,
                                            float* ws) {
  __shared__ float sm[4 * BUF + 80];
  float* B0 = sm;
  float* B1 = sm + BUF;
  float* B2 = sm + 2 * BUF;
  float* B3 = sm + 3 * BUF;
  float* red = sm + 4 * BUF;

  const size_t b = blockIdx.x;
  g2l(B1, X + b * 4096);
  g2l(B0, ws + WS_CI);
  mm64<0>(B0, B1, B2);   // c^-1 @ X
  mm64<0>(B2, B0, B3);   // W = c^-1 X c^-1

  const float t = trace64(B3, red) * (1.0f / 64.0f);
  ew_affine(B3, B3, 1.0f / t, 0.0f);            // normalize: W /= t

  // inverse scaling-and-squaring: K_ROOTS repeated square roots
  float *pY = B3, *pZ = B0, *pT = B1, *pU = B2;
  for (int r = 0; r < K_ROOTS; ++r) ns_sqrt(pY, pZ, pT, pU, NS_ROOT);

  // Gregory/Taylor series: log(R) ~= S(I - S/2 + S^2/3 - S^3/4), S = R - I
  ew_affine(pT, pY, 1.0f, -1.0f);               // S
  ew_affine(pU, pT, -0.25f, 1.0f / 3.0f);       // G1 = I/3 - S/4
  mm64<0>(pT, pU, pY);                          // S@G1
  ew_affine(pY, pY, 1.0f, -0.5f);               // G2 = S@G1 - I/2
  mm64<0>(pT, pY, pZ);                          // S@G2
  ew_affine(pZ, pZ, 1.0f, 1.0f);                // G3 = S@G2 + I
  mm64<0>(pT, pZ, pU);                          // L0 = S@G3
  ew_affine(pU, pU, (float)(1 << K_ROOTS), logf(t));  // L = 2^K L0 + ln(t) I

  g2l(pY, ws + WS_C);                           // c
  mm64<0>(pY, pU, pT);                          // c @ L
  mm64<0>(pT, pY, pZ);                          // P = c L c

  float* part = ws + WS_PART + (size_t)(b & (NPART - 1)) * 4096;
  for (int e = threadIdx.x; e < 4096; e += 128) {
    const int r = e >> 6, c = e & 63;
    atomicAdd(&part[e], pZ[r * LDA + c] * (1.0f / 8192.0f));
  }
}

// single block: m2 = c exp(c^-1 Tm c^-1) c ; c2inv = inv(sqrtm(m2))
__global__ void __launch_bounds__(128) kupdate(float* ws) {
  __shared__ float sm[4 * BUF + 80];
  float* B0 = sm;
  float* B1 = sm + BUF;
  float* B2 = sm + 2 * BUF;
  float* B3 = sm + 3 * BUF;
  float* red = sm + 4 * BUF;

  // reduce the NPART partial tangent sums into LDS
  for (int e = threadIdx.x; e < 4096; e += 128) {
    float s = 0.0f;
    for (int p = 0; p < NPART; ++p) s += ws[WS_PART + p * 4096 + e];
    const int r = e >> 6, c = e & 63;
    B3[r * LDA + c] = s;
  }
  g2l(B0, ws + WS_CI);
  mm64<0>(B0, B3, B1);   // c^-1 @ Tm
  mm64<0>(B1, B0, B2);   // E = c^-1 Tm c^-1

  // matexp: scale by 2^-K_EXP, 8-term Horner Taylor, square K_EXP times
  ew_affine(B3, B2, 1.0f / (float)(1 << K_EXP), 0.0f);  // A
  ew_affine(B1, B3, 0.125f, 1.0f);                      // G = I + A/8
  float* G  = B1;
  float* Gn = B0;
  for (int j = 7; j >= 1; --j) {
    mm64<3>(B3, G, Gn, 1.0f / (float)j);                // G = I + (A@G)/j
    float* t0 = G; G = Gn; Gn = t0;
  }
  for (int i = 0; i < K_EXP; ++i) {
    mm64<0>(G, G, Gn);
    float* t0 = G; G = Gn; Gn = t0;
  }
  g2l(B2, ws + WS_C);
  mm64<0>(B2, G, Gn);    // c @ R
  mm64<0>(Gn, B2, B3);   // m2 = c R c

  const float s2 = trace64(B3, red) * (1.0f / 64.0f);
  ew_affine(B3, B3, 1.0f / s2, 0.0f);
  float *pY = B3, *pZ = B2, *pT = G, *pU = Gn;
  ns_sqrt(pY, pZ, pT, pU, NS_FINAL);
  ew_affine(pT, pZ, 1.0f / sqrtf(s2), 0.0f);    // center_sqrt^-1
  __syncthreads();
  l2g(ws + WS_C2I, pT);
}

// one block per batch element: out_b = c2inv @ X_b @ c2inv
__global__ void __launch_bounds__(128) kout(const float* __restrict__ X,
                                            const float* __restrict__ ws,
                                            float* __restrict__ out) {
  __shared__ float sm[3 * BUF];
  float* B0 = sm;
  float* B1 = sm + BUF;
  float* B2 = sm + 2 * BUF;

  const size_t b = blockIdx.x;
  g2l(B0, ws + WS_C2I);
  g2l(B1, X + b * 4096);
  mm64<0>(B0, B1, B2);
  mm64<0>(B2, B0, B1);
  l2g(out + b * 4096, B1);
}

// ------------------------------- launch ------------------------------------

extern "C" void kernel_launch(void* const* d_in, const int* in_sizes, int n_in,
                              void* d_out, int out_size, void* d_ws,
                              size_t ws_size, hipStream_t stream) {
  (void)in_sizes; (void)n_in; (void)out_size; (void)ws_size;
  const float* X = (const float*)d_in[0];
  float* out = (float*)d_out;
  float* ws  = (float*)d_ws;

  kzero<<<(WS_TOTAL + 255) / 256, 256, 0, stream>>>(ws, WS_TOTAL);
  kmean<<<dim3(32, 64), 128, 0, stream>>>(X, ws);
  kprep<<<1, 128, 0, stream>>>(ws);
  klog<<<8192, 128, 0, stream>>>(X, ws);
  kupdate<<<1, 128, 0, stream>>>(ws);
  kout<<<8192, 128, 0, stream>>>(X, ws, out);
}